// LinearAutoDecoder_47107201303348
// MI455X (gfx1250) — compile-verified
//
#include <hip/hip_runtime.h>
#include <stdint.h>

#define POS_SIZE     63
#define LATENT_SIZE  32
#define DIM          95                     // POS_SIZE + LATENT_SIZE
#define NUM_W_ROWS   384                    // 3 * NUM_CLUSTERS
#define ROW_STRIDE   96                     // words per weight row in LDS (16B aligned)
#define CL_STRIDE    292                    // words per cluster: 4*73 -> 16B aligned, and
                                            // 292*c mod 64 hits all 16 quad-bank groups
#define LDS_WORDS    (128 * CL_STRIDE)      // 37376
#define LDS_BYTES    (LDS_WORDS * 4)        // 149504 (2 blocks fit in 320KB WGP LDS)
#define TILE         256

#if defined(__has_builtin)
#  if __has_builtin(__builtin_amdgcn_global_load_async_to_lds_b32)
#    define HAVE_ASYNC_LDS 1
#  endif
#  if __has_builtin(__builtin_amdgcn_s_wait_asynccnt)
#    define HAVE_WAIT_ASYNC 1
#  endif
#endif

typedef __attribute__((address_space(1))) int* gptr_i32;
typedef __attribute__((address_space(3))) int* lptr_i32;

__device__ __forceinline__ void w_copy(const float* g, float* l) {
#if defined(HAVE_ASYNC_LDS)
  // per-lane async DMA: copies 4 bytes from the global address g to the LDS address l; tracked by ASYNCcnt
  __builtin_amdgcn_global_load_async_to_lds_b32(
      (gptr_i32)(unsigned long long)g,
      (lptr_i32)(unsigned int)(unsigned long long)l,
      0, 0);
#else
  *l = *g;
#endif
}

extern "C" __global__ void __launch_bounds__(256)
lad_kernel(const float* __restrict__ X, const int* __restrict__ ids,
           const float* __restrict__ Wp, const float* __restrict__ Wf,
           float* __restrict__ out, int n) {
  extern __shared__ float wlds[];
  const int tid = threadIdx.x;

  // ---- Stage W = [W_pos | W_feat] into LDS, layout word = c*292 + j*96 + k ----
  for (int e = tid; e < NUM_W_ROWS * POS_SIZE; e += blockDim.x) {
    int r = e / POS_SIZE, k = e - r * POS_SIZE;   // r = 3c + j
    int c = r / 3, j = r - 3 * c;
    w_copy(Wp + e, &wlds[c * CL_STRIDE + j * ROW_STRIDE + k]);
  }
  for (int e = tid; e < NUM_W_ROWS * LATENT_SIZE; e += blockDim.x) {
    int r = e / LATENT_SIZE, k = e - r * LATENT_SIZE;
    int c = r / 3, j = r - 3 * c;
    w_copy(Wf + e, &wlds[c * CL_STRIDE + j * ROW_STRIDE + POS_SIZE + k]);
  }
#if defined(HAVE_ASYNC_LDS)
#  if defined(HAVE_WAIT_ASYNC)
  __builtin_amdgcn_s_wait_asynccnt(0);
#  else
  asm volatile("s_wait_asynccnt 0" ::: "memory");
#  endif
#endif
  __syncthreads();

  // ---- Stream rows: one thread per row, grid-stride over 256-row tiles ----
  const int ntiles = (n + TILE - 1) / TILE;
  for (int t = blockIdx.x; t < ntiles; t += gridDim.x) {
    const int row = t * TILE + tid;
    if (row >= n) continue;                       // no barriers inside this loop

    const float* __restrict__ xr = X + (size_t)row * DIM;
    const float* __restrict__ w  = wlds + (size_t)ids[row] * CL_STRIDE;

    float a0 = 0.f, a1 = 0.f, a2 = 0.f;
#pragma unroll 4
    for (int m = 0; m < 23; ++m) {                // k = 0..91, 16B-aligned LDS quads
      const float4 w0 = *(const float4*)(w + 4 * m);
      const float4 w1 = *(const float4*)(w + ROW_STRIDE + 4 * m);
      const float4 w2 = *(const float4*)(w + 2 * ROW_STRIDE + 4 * m);
      const float x0 = __builtin_nontemporal_load(xr + 4 * m + 0);
      const float x1 = __builtin_nontemporal_load(xr + 4 * m + 1);
      const float x2 = __builtin_nontemporal_load(xr + 4 * m + 2);
      const float x3 = __builtin_nontemporal_load(xr + 4 * m + 3);
      a0 = fmaf(x0, w0.x, a0); a0 = fmaf(x1, w0.y, a0);
      a0 = fmaf(x2, w0.z, a0); a0 = fmaf(x3, w0.w, a0);
      a1 = fmaf(x0, w1.x, a1); a1 = fmaf(x1, w1.y, a1);
      a1 = fmaf(x2, w1.z, a1); a1 = fmaf(x3, w1.w, a1);
      a2 = fmaf(x0, w2.x, a2); a2 = fmaf(x1, w2.y, a2);
      a2 = fmaf(x2, w2.z, a2); a2 = fmaf(x3, w2.w, a2);
    }
#pragma unroll
    for (int k = 92; k < DIM; ++k) {              // tail k = 92..94
      const float x = __builtin_nontemporal_load(xr + k);
      a0 = fmaf(x, w[k], a0);
      a1 = fmaf(x, w[ROW_STRIDE + k], a1);
      a2 = fmaf(x, w[2 * ROW_STRIDE + k], a2);
    }
    __builtin_nontemporal_store(a0, out + (size_t)row * 3 + 0);
    __builtin_nontemporal_store(a1, out + (size_t)row * 3 + 1);
    __builtin_nontemporal_store(a2, out + (size_t)row * 3 + 2);
  }
}

extern "C" void kernel_launch(void* const* d_in, const int* in_sizes, int n_in,
                              void* d_out, int out_size, void* d_ws, size_t ws_size,
                              hipStream_t stream) {
  const float* X   = (const float*)d_in[0];
  const int*   ids = (const int*)d_in[1];
  const float* Wp  = (const float*)d_in[2];
  const float* Wf  = (const float*)d_in[3];
  float*       out = (float*)d_out;
  const int n = in_sizes[1];                      // cluster_ids count == N

  (void)d_ws; (void)ws_size; (void)n_in; (void)out_size;

  // Allow >64KB dynamic LDS (149.5 KB; WGP has 320 KB). Deterministic per call.
  (void)hipFuncSetAttribute((const void*)lad_kernel,
                            hipFuncAttributeMaxDynamicSharedMemorySize, LDS_BYTES);

  const int ntiles = (n + TILE - 1) / TILE;
  int nblocks = ntiles < 1024 ? ntiles : 1024;
  lad_kernel<<<nblocks, 256, LDS_BYTES, stream>>>(X, ids, Wp, Wf, out, n);
}